// AdaptiveFeatureSelector_61134564491481
// MI455X (gfx1250) — compile-verified
//
#include <hip/hip_runtime.h>
#include <hip/hip_bf16.h>

typedef __attribute__((ext_vector_type(2))) float v2f;
typedef __attribute__((ext_vector_type(8))) float v8f;

__device__ __forceinline__ v8f wmma4(v2f a, v2f b, v8f c) {
  // D = A(16x4) * B(4x16) + C, fp32, wave32
  return __builtin_amdgcn_wmma_f32_16x16x4_f32(false, a, false, b, (short)0, c,
                                               false, false);
}

__device__ __forceinline__ float wave_max(float v) {
#pragma unroll
  for (int m = 16; m >= 1; m >>= 1) v = fmaxf(v, __shfl_xor(v, m, 32));
  return v;
}
__device__ __forceinline__ float wave_sum(float v) {
#pragma unroll
  for (int m = 16; m >= 1; m >>= 1) v += __shfl_xor(v, m, 32);
  return v;
}

// ---------------------------------------------------------------------------
// Kernel 1: batch-independent self-attention on Q, then cross-attn q-projection.
// Output: qca[f*8 + h*4 + d] (32 floats) into workspace.
// ---------------------------------------------------------------------------
__global__ __launch_bounds__(32) void afs_precompute(
    const float* __restrict__ Q, const float* __restrict__ sa_w_in,
    const float* __restrict__ sa_b_in, const float* __restrict__ sa_w_out,
    const float* __restrict__ sa_b_out, const float* __restrict__ ca_w_in,
    const float* __restrict__ ca_b_in, float* __restrict__ qca) {
  if (threadIdx.x != 0) return;
  const int F = 4, E = 8;
  float qh[4][8], kh[4][8], vh[4][8];
  for (int f = 0; f < F; ++f)
    for (int j = 0; j < 8; ++j) {
      float sq = sa_b_in[j], sk = sa_b_in[8 + j], sv = sa_b_in[16 + j];
      for (int e = 0; e < E; ++e) {
        float x = Q[f * E + e];
        sq += x * sa_w_in[j * E + e];
        sk += x * sa_w_in[(8 + j) * E + e];
        sv += x * sa_w_in[(16 + j) * E + e];
      }
      qh[f][j] = sq; kh[f][j] = sk; vh[f][j] = sv;
    }
  float ctx[4][8];
  for (int h = 0; h < 2; ++h)
    for (int f = 0; f < F; ++f) {
      float sc[4], m = -3.0e38f;
      for (int g = 0; g < F; ++g) {
        float s = 0.f;
        for (int d = 0; d < 4; ++d) s += qh[f][h * 4 + d] * kh[g][h * 4 + d];
        s *= 0.5f;  // 1/sqrt(Dh=4)
        sc[g] = s; m = fmaxf(m, s);
      }
      float den = 0.f;
      for (int g = 0; g < F; ++g) { sc[g] = __expf(sc[g] - m); den += sc[g]; }
      for (int d = 0; d < 4; ++d) {
        float a = 0.f;
        for (int g = 0; g < F; ++g) a += sc[g] * vh[g][h * 4 + d];
        ctx[f][h * 4 + d] = a / den;
      }
    }
  for (int f = 0; f < F; ++f) {
    float qu[8];
    for (int e = 0; e < E; ++e) {
      float s = sa_b_out[e];
      for (int j = 0; j < 8; ++j) s += ctx[f][j] * sa_w_out[e * E + j];
      qu[e] = s;
    }
    for (int j = 0; j < 8; ++j) {
      float s = ca_b_in[j];
      for (int e = 0; e < E; ++e) s += qu[e] * ca_w_in[j * E + e];
      qca[f * 8 + j] = s;
    }
  }
}

// ---------------------------------------------------------------------------
// Kernel 2: one block per batch element. 256 threads = 8 wave32s.
// ---------------------------------------------------------------------------
__global__ __launch_bounds__(256) void afs_main(
    const float* __restrict__ x, const float* __restrict__ ca_w_in,
    const float* __restrict__ ca_b_in, const float* __restrict__ ca_w_out,
    const float* __restrict__ ca_b_out, const float* __restrict__ r_w1,
    const float* __restrict__ r_b1, const float* __restrict__ r_w2,
    const float* __restrict__ r_b2, const float* __restrict__ qca,
    float* __restrict__ out, float* __restrict__ sim) {
  __shared__ float xB[512 * 9];    // x[s][e], padded stride 9
  __shared__ float vhB[512 * 9];   // vh[s][j], padded stride 9
  __shared__ float scB[8 * 520];   // scores/attn rows, padded stride 520
  __shared__ float khT[8 * 144];   // per-wave kh transpose tile [16][9]
  __shared__ float wkvB[128];      // [e][n]: n<8 -> wk[n][e], n>=8 -> wv[n-8][e]
  __shared__ float sB[128];        // scores B fragment [k][n] (zero padded)
  __shared__ float simB[128];      // sim B fragment [e][n] (cols>=4 zero)
  __shared__ float biasKV[16];
  __shared__ float woutS[64];
  __shared__ float boutS[8];
  __shared__ float rw1S[512];
  __shared__ float rb1S[16];
  __shared__ float rw2S[16];
  __shared__ float ctxS[32];
  __shared__ float ffS[32];

  const int tid = threadIdx.x;
  const int b = blockIdx.x;
  const int wave = tid >> 5, lane = tid & 31;
  const int half = lane >> 4, l = lane & 15;

  // ---- Phase 1: stage x (b128 loads) and small operands into LDS ----
  const float* xg = x + (size_t)b * 4096;
  {
    const float4* xg4 = (const float4*)xg;
#pragma unroll
    for (int k = 0; k < 4; ++k) {
      int i = tid + k * 256;          // 1024 float4s total
      float4 w = xg4[i];
      float* d = &xB[(i >> 1) * 9 + (i & 1) * 4];
      d[0] = w.x; d[1] = w.y; d[2] = w.z; d[3] = w.w;
    }
  }
  if (tid < 128) {
    int e = tid >> 4, n = tid & 15;
    wkvB[tid] = ca_w_in[(8 + n) * 8 + e];  // rows 8..23 of packed in_proj
    int k = e, h = k >> 2, d = k & 3;
    float v = 0.f;
    if (h == 0) { if (n < 4) v = qca[n * 8 + d]; }
    else        { if (n >= 4 && n < 8) v = qca[(n - 4) * 8 + 4 + d]; }
    sB[tid] = v;
    simB[tid] = 0.f;
  }
  if (tid < 16) biasKV[tid] = ca_b_in[8 + tid];
  if (tid < 64) woutS[tid] = ca_w_out[tid];
  if (tid < 8)  boutS[tid] = ca_b_out[tid];
  for (int i = tid; i < 512; i += 256) rw1S[i] = r_w1[i];
  if (tid < 16) { rb1S[tid] = r_b1[tid]; rw2S[tid] = r_w2[tid]; }
  __syncthreads();

  // Loop-invariant per-lane pieces
  const int klo = 2 * half;            // K offset within a 4-chunk
  float* const khw = &khT[wave * 144]; // this wave's kh transpose tile
  const float biasL = biasKV[l];
  const v2f wb0 = {wkvB[klo * 16 + l], wkvB[(klo + 1) * 16 + l]};
  const v2f wb1 = {wkvB[(4 + klo) * 16 + l], wkvB[(5 + klo) * 16 + l]};
  const v2f qb0 = {sB[klo * 16 + l], sB[(klo + 1) * 16 + l]};
  const v2f qb1 = {sB[(4 + klo) * 16 + l], sB[(5 + klo) * 16 + l]};

  // ---- Phase 2+3 fused: KV projection WMMA + scores WMMA per 16-row tile ----
  for (int t = wave; t < 32; t += 8) {
    const int s0 = t * 16;
    const float* xr = &xB[(s0 + l) * 9];
    v2f a0, a1;
    a0.x = xr[klo];     a0.y = xr[klo + 1];
    a1.x = xr[4 + klo]; a1.y = xr[4 + klo + 1];
    v8f c = {};
    c = wmma4(a0, wb0, c);
    c = wmma4(a1, wb1, c);
    // Branchless store: kh cols (l<8) to per-wave transpose tile, vh cols to
    // block LDS. Both destinations have row-stride 9.
    {
      float* khb = khw + l;                  // kh[m][l], stride 9
      float* vhb = &vhB[s0 * 9 + (l - 8)];   // vh[s0+m][l-8], stride 9
      float* base = (l < 8) ? khb : vhb;
      base += (8 * half) * 9;                // fold m = r + 8*half
#pragma unroll
      for (int r = 0; r < 8; ++r) base[r * 9] = c[r] + biasL;
    }
    asm volatile("s_wait_dscnt 0" ::: "memory");  // wave-local LDS transpose
    v2f sa0, sa1;
    const float* kr = &khw[l * 9];
    sa0.x = kr[klo];     sa0.y = kr[klo + 1];
    sa1.x = kr[4 + klo]; sa1.y = kr[4 + klo + 1];
    v8f sc = {};
    sc = wmma4(sa0, qb0, sc);
    sc = wmma4(sa1, qb1, sc);
    if (l < 8) {
      float* dst = &scB[l * 520 + s0 + 8 * half];
#pragma unroll
      for (int r = 0; r < 8; ++r) dst[r] = 0.5f * sc[r];
    }
  }
  __syncthreads();

  // ---- Phase 4: softmax over S for 8 (h,f) rows, one wave each ----
  {
    float vals[16], m = -3.0e38f;
#pragma unroll
    for (int i = 0; i < 16; ++i) {
      vals[i] = scB[wave * 520 + lane + 32 * i];
      m = fmaxf(m, vals[i]);
    }
    m = wave_max(m);
    float s = 0.f;
#pragma unroll
    for (int i = 0; i < 16; ++i) { vals[i] = __expf(vals[i] - m); s += vals[i]; }
    s = wave_sum(s);
    float inv = 1.0f / s;
#pragma unroll
    for (int i = 0; i < 16; ++i) scB[wave * 520 + lane + 32 * i] = vals[i] * inv;
  }
  __syncthreads();

  // ---- Phase 5: ctx[f][h*4+d] = sum_s attn * vh ----
  {
    const int h = wave >> 2, f = wave & 3;
    float a0 = 0.f, a1 = 0.f, a2 = 0.f, a3 = 0.f;
    for (int i = lane; i < 512; i += 32) {
      float a = scB[wave * 520 + i];
      const float* vr = &vhB[i * 9 + h * 4];
      a0 += a * vr[0]; a1 += a * vr[1]; a2 += a * vr[2]; a3 += a * vr[3];
    }
    a0 = wave_sum(a0); a1 = wave_sum(a1); a2 = wave_sum(a2); a3 = wave_sum(a3);
    if (lane == 0) {
      ctxS[f * 8 + h * 4 + 0] = a0; ctxS[f * 8 + h * 4 + 1] = a1;
      ctxS[f * 8 + h * 4 + 2] = a2; ctxS[f * 8 + h * 4 + 3] = a3;
    }
  }
  __syncthreads();

  // ---- Phase 6: out-projection -> Q_cross, packed as sim B fragment ----
  if (wave == 0) {
    int f = lane >> 3, e = lane & 7;
    float q = boutS[e];
#pragma unroll
    for (int j = 0; j < 8; ++j) q += ctxS[f * 8 + j] * woutS[e * 8 + j];
    simB[e * 16 + f] = q;
  }
  __syncthreads();

  // ---- Phase 7: similarity WMMA into LDS ----
  {
    const v2f cb0 = {simB[klo * 16 + l], simB[(klo + 1) * 16 + l]};
    const v2f cb1 = {simB[(4 + klo) * 16 + l], simB[(5 + klo) * 16 + l]};
    for (int t = wave; t < 32; t += 8) {
      const int s0 = t * 16;
      const float* xr = &xB[(s0 + l) * 9];
      v2f a0, a1;
      a0.x = xr[klo];     a0.y = xr[klo + 1];
      a1.x = xr[4 + klo]; a1.y = xr[4 + klo + 1];
      v8f c = {};
      c = wmma4(a0, cb0, c);
      c = wmma4(a1, cb1, c);
      if (l < 4) {
        float* dst = &scB[l * 520 + s0 + 8 * half];
#pragma unroll
        for (int r = 0; r < 8; ++r) dst[r] = c[r];
      }
    }
  }
  __syncthreads();

  // ---- Phase 7b: coalesced b128 store of similarity to global ----
  {
    float4* simg4 = (float4*)(sim + (size_t)b * 2048);
#pragma unroll
    for (int k = 0; k < 2; ++k) {
      int i = tid + k * 256;            // 512 float4s total
      int f = i >> 7, s = (i << 2) & 511;
      const float* src = &scB[f * 520 + s];
      float4 w;
      w.x = src[0]; w.y = src[1]; w.z = src[2]; w.w = src[3];
      simg4[i] = w;
    }
  }
  __syncthreads();

  // ---- Phase 8: softmax over S for the 4 similarity rows ----
  if (wave < 4) {
    float vals[16], m = -3.0e38f;
#pragma unroll
    for (int i = 0; i < 16; ++i) {
      vals[i] = scB[wave * 520 + lane + 32 * i];
      m = fmaxf(m, vals[i]);
    }
    m = wave_max(m);
    float s = 0.f;
#pragma unroll
    for (int i = 0; i < 16; ++i) { vals[i] = __expf(vals[i] - m); s += vals[i]; }
    s = wave_sum(s);
    float inv = 1.0f / s;
#pragma unroll
    for (int i = 0; i < 16; ++i) scB[wave * 520 + lane + 32 * i] = vals[i] * inv;
  }
  __syncthreads();

  // ---- Phase 9: final_factors[f][e] = sum_s attnW * x ----
  if (wave < 4) {
    float acc[8] = {0.f, 0.f, 0.f, 0.f, 0.f, 0.f, 0.f, 0.f};
    for (int i = lane; i < 512; i += 32) {
      float a = scB[wave * 520 + i];
      const float* xr = &xB[i * 9];
#pragma unroll
      for (int e = 0; e < 8; ++e) acc[e] += a * xr[e];
    }
#pragma unroll
    for (int e = 0; e < 8; ++e) {
      float s = wave_sum(acc[e]);
      if (lane == 0) ffS[wave * 8 + e] = s;
    }
  }
  __syncthreads();

  // ---- Phase 10: tiny MLP head ----
  if (wave == 0) {
    float hj = 0.f;
    if (lane < 16) {
      hj = rb1S[lane];
#pragma unroll
      for (int i = 0; i < 32; ++i) hj += ffS[i] * rw1S[lane * 32 + i];
      hj = fmaxf(hj, 0.f) * rw2S[lane];
    }
    float s = wave_sum(hj);
    if (lane == 0) out[b] = s + r_b2[0];
  }
}

extern "C" void kernel_launch(void* const* d_in, const int* in_sizes, int n_in,
                              void* d_out, int out_size, void* d_ws,
                              size_t ws_size, hipStream_t stream) {
  const float* x        = (const float*)d_in[0];
  const float* Q        = (const float*)d_in[1];
  const float* sa_w_in  = (const float*)d_in[2];
  const float* sa_b_in  = (const float*)d_in[3];
  const float* sa_w_out = (const float*)d_in[4];
  const float* sa_b_out = (const float*)d_in[5];
  const float* ca_w_in  = (const float*)d_in[6];
  const float* ca_b_in  = (const float*)d_in[7];
  const float* ca_w_out = (const float*)d_in[8];
  const float* ca_b_out = (const float*)d_in[9];
  const float* r_w1     = (const float*)d_in[10];
  const float* r_b1     = (const float*)d_in[11];
  const float* r_w2     = (const float*)d_in[12];
  const float* r_b2     = (const float*)d_in[13];

  const int B = in_sizes[0] / (512 * 8);
  float* qca = (float*)d_ws;                 // 32 floats of scratch
  float* out = (float*)d_out;                // [B,1] first
  float* sim = out + B;                      // then [B,4,512]

  afs_precompute<<<1, 32, 0, stream>>>(Q, sa_w_in, sa_b_in, sa_w_out, sa_b_out,
                                       ca_w_in, ca_b_in, qca);
  afs_main<<<B, 256, 0, stream>>>(x, ca_w_in, ca_b_in, ca_w_out, ca_b_out,
                                  r_w1, r_b1, r_w2, r_b2, qca, out, sim);
}